// YOLOvx_45303315038383
// MI455X (gfx1250) — compile-verified
//
#include <hip/hip_runtime.h>
#include <cstdint>

#ifndef __has_builtin
#define __has_builtin(x) 0
#endif

#define NUM_CLASSES 80
#define TOPK 1000
#define NTOT 3000                 // 3 levels * 1000
#define NW 94                     // ceil(3008/32) words per suppression row
#define NBINS 4096
#define CAP 8192                  // per-level candidate cap (power of 2 for bitonic)
#define SORTN 4096                // merged sort size (pad 3000 -> 4096)
#define CONF 0.05f
#define NMS_T 0.6f
#define CLS_OFF 100000.0f

typedef float v2f __attribute__((ext_vector_type(2)));
typedef float v8f __attribute__((ext_vector_type(8)));

// ---- gfx1250 feature probes (guarded: fallback keeps compile green) ----
#if defined(__HIP_DEVICE_COMPILE__) && __has_builtin(__builtin_amdgcn_wmma_f32_16x16x4_f32)
#define HAS_WMMA4 1
#endif
#if defined(__HIP_DEVICE_COMPILE__) && __has_builtin(__builtin_amdgcn_global_load_async_to_lds_b32)
#define HAS_ASYNC 1
#endif
#if defined(__HIP_DEVICE_COMPILE__) && __has_builtin(__builtin_amdgcn_s_wait_asynccnt)
#define WAIT_ASYNC(n) __builtin_amdgcn_s_wait_asynccnt(n)
#elif defined(__HIP_DEVICE_COMPILE__)
#define WAIT_ASYNC(n) asm volatile("s_wait_asynccnt %0" ::"i"(n) : "memory")
#else
#define WAIT_ASYNC(n) do { } while (0)
#endif

#if HAS_ASYNC
__device__ __forceinline__ void g2l_async_b32(const void* g, void* l) {
  __builtin_amdgcn_global_load_async_to_lds_b32(
      (__attribute__((address_space(1))) int*)(uintptr_t)g,
      (__attribute__((address_space(3))) int*)(uintptr_t)l, 0, 0);
}
#endif

// ---- workspace layout (~1.52 MB) ----
struct Ws {
  int hist[3 * NBINS];
  int thresh[4];
  int counter[4];
  float candS[3 * CAP];
  int candI[3 * CAP];
  float ubox[NTOT * 4];   // unsorted merged (level-major, rank order)
  int ulab[NTOT];
  float uscore[NTOT];     // masked score: s if s>CONF else -1
  float sbox[NTOT * 4];   // globally sorted
  int slab[NTOT];
  float sscore[NTOT];
  unsigned sup[NTOT * NW]; // suppression bitmask matrix
};

__device__ __forceinline__ float sigm(float x) { return 1.f / (1.f + expf(-x)); }

// total order: score desc, index asc (matches lax.top_k / stable argsort(-s))
__device__ __forceinline__ bool before_(float s1, int i1, float s2, int i2) {
  return (s1 > s2) || (s1 == s2 && i1 < i2);
}

__device__ void bitonic(float* ks, int* vi, int N) {
  for (int k = 2; k <= N; k <<= 1) {
    for (int j = k >> 1; j > 0; j >>= 1) {
      for (int i = threadIdx.x; i < N; i += blockDim.x) {
        int p = i ^ j;
        if (p > i) {
          bool up = (i & k) == 0;
          bool sw = up ? before_(ks[p], vi[p], ks[i], vi[i])
                       : before_(ks[i], vi[i], ks[p], vi[p]);
          if (sw) {
            float ts = ks[i]; ks[i] = ks[p]; ks[p] = ts;
            int tv = vi[i]; vi[i] = vi[p]; vi[p] = tv;
          }
        }
      }
      __syncthreads();
    }
  }
}

// ---- k0: zero counters/histograms (ws is poisoned once, never re-poisoned) ----
__global__ void zero_k(int* __restrict__ hist, int* __restrict__ counter) {
  int t = blockIdx.x * blockDim.x + threadIdx.x;
  if (t < 3 * NBINS) hist[t] = 0;
  if (t < 3) counter[t] = 0;
}

// ---- k1: streaming score histogram (b128 cls loads; one block-local hist) ----
__global__ void hist_k(const float* __restrict__ obj, const float* __restrict__ cls,
                       int n4, int* __restrict__ hist) {
  __shared__ int lh[NBINS];
  for (int b = threadIdx.x; b < NBINS; b += blockDim.x) lh[b] = 0;
  __syncthreads();
  int t = blockIdx.x * blockDim.x + threadIdx.x;
  if (t < n4) {
    float4 c4 = ((const float4*)cls)[t];           // 80 % 4 == 0 -> one anchor per float4
    float so = sigm(obj[(t * 4) / NUM_CLASSES]);
    float v[4] = {c4.x, c4.y, c4.z, c4.w};
#pragma unroll
    for (int k = 0; k < 4; ++k) {
      float s = sqrtf(so * sigm(v[k]));
      int b = (int)(s * (float)NBINS);
      b = b < 0 ? 0 : (b > NBINS - 1 ? NBINS - 1 : b);
      atomicAdd(&lh[b], 1);
    }
  }
  __syncthreads();
  for (int b = threadIdx.x; b < NBINS; b += blockDim.x) {
    int v = lh[b];
    if (v) atomicAdd(&hist[b], v);
  }
}

// ---- k2: per-level cutoff bin so that count(bin >= thresh) >= TOPK ----
__global__ void thresh_k(const int* __restrict__ hist, int* __restrict__ thresh) {
  int t = threadIdx.x;
  if (t < 3) {
    int cum = 0, res = 0;
    for (int b = NBINS - 1; b >= 0; --b) {
      cum += hist[t * NBINS + b];
      if (cum >= TOPK) { res = b; break; }
    }
    thresh[t] = res;
  }
}

// ---- k3: compact candidates above cutoff (order fixed later by full sort) ----
__global__ void compact_k(const float* __restrict__ obj, const float* __restrict__ cls,
                          int n4, int lvl, const int* __restrict__ thresh,
                          float* __restrict__ candS, int* __restrict__ candI,
                          int* __restrict__ counter) {
  int t = blockIdx.x * blockDim.x + threadIdx.x;
  if (t >= n4) return;
  int th = thresh[lvl];
  float4 c4 = ((const float4*)cls)[t];
  int base = t * 4;
  float so = sigm(obj[base / NUM_CLASSES]);
  float v[4] = {c4.x, c4.y, c4.z, c4.w};
#pragma unroll
  for (int k = 0; k < 4; ++k) {
    float s = sqrtf(so * sigm(v[k]));
    int b = (int)(s * (float)NBINS);
    b = b < 0 ? 0 : (b > NBINS - 1 ? NBINS - 1 : b);
    if (b >= th) {
      int pos = atomicAdd(&counter[lvl], 1);
      if (pos < CAP) {
        candS[lvl * CAP + pos] = s;
        candI[lvl * CAP + pos] = base + k;
      }
    }
  }
}

// ---- k4: per-level exact top-1000 via full bitonic sort of candidates ----
__global__ __launch_bounds__(1024) void topk_k(
    const float* __restrict__ candS, const int* __restrict__ candI,
    const int* __restrict__ counter, const float* __restrict__ box0,
    const float* __restrict__ box1, const float* __restrict__ box2,
    float* __restrict__ ubox, int* __restrict__ ulab, float* __restrict__ uscore) {
  __shared__ float ks[CAP];
  __shared__ int vi[CAP];
  const int lvl = blockIdx.x;
  int cnt = counter[lvl];
  cnt = cnt > CAP ? CAP : cnt;
  for (int i = threadIdx.x; i < CAP; i += blockDim.x) {
    if (i < cnt) { ks[i] = candS[lvl * CAP + i]; vi[i] = candI[lvl * CAP + i]; }
    else         { ks[i] = -1e30f;               vi[i] = 0x7FFFFFFF; }
  }
  __syncthreads();
  bitonic(ks, vi, CAP);
  const float* bp = (lvl == 0) ? box0 : (lvl == 1) ? box1 : box2;
  for (int r = threadIdx.x; r < TOPK; r += blockDim.x) {
    float s = ks[r];
    int idx = vi[r];
    int anchor = 0, lab = 0;
    if (idx != 0x7FFFFFFF) { anchor = idx / NUM_CLASSES; lab = idx % NUM_CLASSES; }
    else s = -1e30f;
    int g = lvl * TOPK + r;
    ubox[g * 4 + 0] = bp[anchor * 4 + 0];
    ubox[g * 4 + 1] = bp[anchor * 4 + 1];
    ubox[g * 4 + 2] = bp[anchor * 4 + 2];
    ubox[g * 4 + 3] = bp[anchor * 4 + 3];
    ulab[g] = lab;
    uscore[g] = (s > CONF) ? s : -1.0f;   // masked score (reference: where(valid, s, -1))
  }
}

// ---- k5: global sort of 3000 by (masked score desc, position asc) ----
__global__ __launch_bounds__(1024) void msort_k(
    const float* __restrict__ uscore, const float* __restrict__ ubox,
    const int* __restrict__ ulab, float* __restrict__ sbox, int* __restrict__ slab,
    float* __restrict__ sscore) {
  __shared__ float ks[SORTN];
  __shared__ int vi[SORTN];
  for (int i = threadIdx.x; i < SORTN; i += blockDim.x) {
    ks[i] = (i < NTOT) ? uscore[i] : -1e30f;
    vi[i] = i;
  }
  __syncthreads();
  bitonic(ks, vi, SORTN);
  for (int r = threadIdx.x; r < NTOT; r += blockDim.x) {
    int src = vi[r];
    sscore[r] = ks[r];
    slab[r] = ulab[src];
    sbox[r * 4 + 0] = ubox[src * 4 + 0];
    sbox[r * 4 + 1] = ubox[src * 4 + 1];
    sbox[r * 4 + 2] = ubox[src * 4 + 2];
    sbox[r * 4 + 3] = ubox[src * 4 + 3];
  }
}

// ---- k6: suppression matrix, one wave per 16x32 tile ----
// WMMA f32 16x16x4 computes the exact outer-sum area_i + area_j per tile
// (products x1.0 are exact -> bit-identical to scalar f32 adds).
__global__ __launch_bounds__(32) void supmat_k(const float* __restrict__ sbox,
                                               const int* __restrict__ slab,
                                               unsigned* __restrict__ sup) {
  __shared__ float rb[16][4];
  __shared__ float ra[16];
  __shared__ float cb[32][4];
  __shared__ float ca[32];
  const int lane = threadIdx.x;
  const int row0 = blockIdx.y << 4;
  const int colw = blockIdx.x;
  const int col0 = colw << 5;

  { // stage 32 column boxes (class-offset applied); OOB -> degenerate zero box
    int j = col0 + lane;
    float x1 = 0.f, y1 = 0.f, x2 = 0.f, y2 = 0.f;
    if (j < NTOT) {
      float off = (float)slab[j] * CLS_OFF;
      x1 = sbox[j * 4 + 0] + off; y1 = sbox[j * 4 + 1] + off;
      x2 = sbox[j * 4 + 2] + off; y2 = sbox[j * 4 + 3] + off;
    }
    cb[lane][0] = x1; cb[lane][1] = y1; cb[lane][2] = x2; cb[lane][3] = y2;
    ca[lane] = fmaxf(x2 - x1, 0.f) * fmaxf(y2 - y1, 0.f);
  }
  if (lane < 16) { // stage 16 row boxes
    int i = row0 + lane;
    float x1 = 0.f, y1 = 0.f, x2 = 0.f, y2 = 0.f;
    if (i < NTOT) {
      float off = (float)slab[i] * CLS_OFF;
      x1 = sbox[i * 4 + 0] + off; y1 = sbox[i * 4 + 1] + off;
      x2 = sbox[i * 4 + 2] + off; y2 = sbox[i * 4 + 3] + off;
    }
    rb[lane][0] = x1; rb[lane][1] = y1; rb[lane][2] = x2; rb[lane][3] = y2;
    ra[lane] = fmaxf(x2 - x1, 0.f) * fmaxf(y2 - y1, 0.f);
  }
  __syncthreads();

  float as0[8], as1[8]; // areaSum for D-reg r: (M = r + 8*(lane>>4), N = lane&15 [+16])
#if HAS_WMMA4
  {
    // A(16x4): A[M][0]=area_row[M], A[M][1]=1 ; B(4x16): B[0][N]=1, B[1][N]=area_col[N]
    v2f a, b0, b1;
    a.x = (lane < 16) ? ra[lane] : 0.f;
    a.y = (lane < 16) ? 1.f : 0.f;
    b0.x = (lane < 16) ? 1.f : 0.f;
    b0.y = (lane < 16) ? ca[lane] : 0.f;
    b1.x = b0.x;
    b1.y = (lane < 16) ? ca[lane + 16] : 0.f;
    v8f c0 = {0.f, 0.f, 0.f, 0.f, 0.f, 0.f, 0.f, 0.f};
    v8f d0 = __builtin_amdgcn_wmma_f32_16x16x4_f32(false, a, false, b0, (short)0, c0, false, false);
    v8f d1 = __builtin_amdgcn_wmma_f32_16x16x4_f32(false, a, false, b1, (short)0, c0, false, false);
    union { v8f v; float f[8]; } u0, u1;
    u0.v = d0; u1.v = d1;
#pragma unroll
    for (int r = 0; r < 8; ++r) { as0[r] = u0.f[r]; as1[r] = u1.f[r]; }
  }
#else
  {
    const int Mb = (lane >> 4) << 3;
    const int Nc = lane & 15;
#pragma unroll
    for (int r = 0; r < 8; ++r) {
      as0[r] = ra[r + Mb] + ca[Nc];
      as1[r] = ra[r + Mb] + ca[Nc + 16];
    }
  }
#endif

  unsigned myword = 0;
  const int Mb = (lane >> 4) << 3;
  const int Nc = lane & 15;
#pragma unroll
  for (int r = 0; r < 8; ++r) {
    const int M = r + Mb;
    float ix1 = fmaxf(rb[M][0], cb[Nc][0]);
    float iy1 = fmaxf(rb[M][1], cb[Nc][1]);
    float ix2 = fminf(rb[M][2], cb[Nc][2]);
    float iy2 = fminf(rb[M][3], cb[Nc][3]);
    float inter0 = fmaxf(ix2 - ix1, 0.f) * fmaxf(iy2 - iy1, 0.f);
    bool bit0 = (inter0 / (as0[r] - inter0 + 1e-10f)) > NMS_T;

    ix1 = fmaxf(rb[M][0], cb[Nc + 16][0]);
    iy1 = fmaxf(rb[M][1], cb[Nc + 16][1]);
    ix2 = fminf(rb[M][2], cb[Nc + 16][2]);
    iy2 = fminf(rb[M][3], cb[Nc + 16][3]);
    float inter1 = fmaxf(ix2 - ix1, 0.f) * fmaxf(iy2 - iy1, 0.f);
    bool bit1 = (inter1 / (as1[r] - inter1 + 1e-10f)) > NMS_T;

#if defined(__HIP_DEVICE_COMPILE__) && __has_builtin(__builtin_amdgcn_ballot_w32)
    unsigned m0 = __builtin_amdgcn_ballot_w32(bit0);
    unsigned m1 = __builtin_amdgcn_ballot_w32(bit1);
#else
    unsigned m0 = (unsigned)__ballot(bit0);
    unsigned m1 = (unsigned)__ballot(bit1);
#endif
    // D layout: ballot bits[15:0] = row M=r, bits[31:16] = row M=r+8 (same 16 cols)
    if (lane == r)     myword = (m0 & 0xFFFFu) | ((m1 & 0xFFFFu) << 16);
    if (lane == r + 8) myword = (m0 >> 16) | (m1 & 0xFFFF0000u);
  }
  if (lane < 16) {
    int i = row0 + lane;
    if (i < NTOT) sup[i * NW + colw] = myword;
  }
}

// ---- k7: serial greedy-NMS reduce + final output ----
// Async global->LDS ping-pong prefetch of row i+1 while row i is consumed.
__global__ __launch_bounds__(128) void nms_reduce_k(
    const unsigned* __restrict__ sup, const float* __restrict__ sbox,
    const int* __restrict__ slab, const float* __restrict__ sscore,
    float* __restrict__ outf) {
  __shared__ unsigned curbit;
  __shared__ unsigned kw[NW];
#if HAS_ASYNC
  __shared__ unsigned rowbuf[2][NW];
#endif
  const int w = threadIdx.x;
  unsigned keep = 0;
  if (w < NW) {
    for (int b = 0; b < 32; ++b) {
      int j = (w << 5) + b;
      if (j < NTOT && sscore[j] > CONF) keep |= (1u << b);  // keep init = valid
    }
  }
#if HAS_ASYNC
  if (w < NW) g2l_async_b32(&sup[w], &rowbuf[0][w]);
  WAIT_ASYNC(0);
#endif
  __syncthreads();
  for (int i = 0; i < NTOT; ++i) {
#if HAS_ASYNC
    const int cur = i & 1;
    if ((i + 1) < NTOT) {
      if (w < NW) g2l_async_b32(&sup[(i + 1) * NW + w], &rowbuf[cur ^ 1][w]);
      WAIT_ASYNC(1);  // oldest outstanding (row i) complete; row i+1 may remain in flight
    } else {
      WAIT_ASYNC(0);  // nothing new issued: drain so row i is guaranteed resident
    }
#endif
    if (w == (i >> 5)) curbit = (keep >> (i & 31)) & 1u;
    __syncthreads();
    if (curbit && w < NW) {
#if HAS_ASYNC
      unsigned s = rowbuf[cur][w];
#else
      unsigned s = sup[i * NW + w];
#endif
      const int wi = i >> 5;
      if (w < wi) s = 0u;                       // only j > i may be suppressed
      else if (w == wi) {
        unsigned m = ((i & 31) == 31) ? 0xFFFFFFFFu : ((1u << ((i & 31) + 1)) - 1u);
        s &= ~m;
      }
      keep &= ~s;
    }
    __syncthreads();
  }
  if (w < NW) kw[w] = keep;
  __syncthreads();
  for (int r = w; r < NTOT; r += blockDim.x) {
    unsigned b = (kw[r >> 5] >> (r & 31)) & 1u;
    float s = sscore[r];
    outf[r * 5 + 0] = sbox[r * 4 + 0];
    outf[r * 5 + 1] = sbox[r * 4 + 1];
    outf[r * 5 + 2] = sbox[r * 4 + 2];
    outf[r * 5 + 3] = sbox[r * 4 + 3];
    outf[r * 5 + 4] = b ? s : 0.0f;
    outf[NTOT * 5 + r] = (float)slab[r];
  }
}

extern "C" void kernel_launch(void* const* d_in, const int* in_sizes, int n_in,
                              void* d_out, int out_size, void* d_ws, size_t ws_size,
                              hipStream_t stream) {
  (void)n_in; (void)out_size; (void)ws_size;
  const float* obj[3]; const float* cls[3]; const float* box[3]; int hw[3];
  for (int l = 0; l < 3; ++l) {
    obj[l] = (const float*)d_in[3 * l + 0];
    cls[l] = (const float*)d_in[3 * l + 1];
    box[l] = (const float*)d_in[3 * l + 2];
    hw[l] = in_sizes[3 * l + 0];            // obj flat size == hw
  }
  Ws* ws = (Ws*)d_ws;

  zero_k<<<(3 * NBINS + 255) / 256, 256, 0, stream>>>(ws->hist, ws->counter);
  for (int l = 0; l < 3; ++l) {
    int n4 = hw[l] * (NUM_CLASSES / 4);
    hist_k<<<(n4 + 255) / 256, 256, 0, stream>>>(obj[l], cls[l], n4, ws->hist + l * NBINS);
  }
  thresh_k<<<1, 32, 0, stream>>>(ws->hist, ws->thresh);
  for (int l = 0; l < 3; ++l) {
    int n4 = hw[l] * (NUM_CLASSES / 4);
    compact_k<<<(n4 + 255) / 256, 256, 0, stream>>>(obj[l], cls[l], n4, l, ws->thresh,
                                                    ws->candS, ws->candI, ws->counter);
  }
  topk_k<<<3, 1024, 0, stream>>>(ws->candS, ws->candI, ws->counter, box[0], box[1], box[2],
                                 ws->ubox, ws->ulab, ws->uscore);
  msort_k<<<1, 1024, 0, stream>>>(ws->uscore, ws->ubox, ws->ulab, ws->sbox, ws->slab,
                                  ws->sscore);
  dim3 g7(NW, (NTOT + 15) / 16);
  supmat_k<<<g7, 32, 0, stream>>>(ws->sbox, ws->slab, ws->sup);
  nms_reduce_k<<<1, 128, 0, stream>>>(ws->sup, ws->sbox, ws->slab, ws->sscore, (float*)d_out);
}